// GIN_74612171866510
// MI455X (gfx1250) — compile-verified
//
#include <hip/hip_runtime.h>
#include <hip/hip_bf16.h>

#define NNODES 50000
#define NEDGES 800000
#define NHID   64
#define NGRAPH 64
#define BN_EPS 1e-5f
#define NF (NNODES * NHID)

typedef __attribute__((ext_vector_type(2))) float v2f;
typedef __attribute__((ext_vector_type(8))) float v8f;

// ---------------------------------------------------------------- utilities
__global__ void zero_f32(float* __restrict__ p, int n) {
    int i = blockIdx.x * blockDim.x + threadIdx.x;
    for (; i < n; i += gridDim.x * blockDim.x) p[i] = 0.0f;
}

__global__ void copy_f32(const float* __restrict__ s, float* __restrict__ d, int n) {
    int i = blockIdx.x * blockDim.x + threadIdx.x;
    if (i < n) d[i] = s[i];
}

// ------------------------------------------------- edge gather / scatter-add
// t[dst] += h[src] for every edge; t was pre-initialized with h (self term).
__global__ void edge_scatter(const float* __restrict__ h,
                             const int* __restrict__ src,
                             const int* __restrict__ dst,
                             float* __restrict__ t) {
    int tid = blockIdx.x * blockDim.x + threadIdx.x;
    int e = tid >> 4;            // 16 threads per edge
    int c = (tid & 15) * 4;      // 4 floats each
    if (e >= NEDGES) return;
    int s = src[e], d = dst[e];
    const float4 v = *(const float4*)(h + (size_t)s * NHID + c);
    float* o = t + (size_t)d * NHID + c;
    atomicAdd(o + 0, v.x);
    atomicAdd(o + 1, v.y);
    atomicAdd(o + 2, v.z);
    atomicAdd(o + 3, v.w);
}

// --------------------------------------------------- WMMA fp32 GEMM kernels
// Each wave computes one 16x16 output tile of  out = A[N x 64] @ W[64 x 64],
// accumulating K=64 in sixteen V_WMMA_F32_16X16X4_F32 steps.
#define ROW_TILES (NNODES / 16)      /* 3125 exact */
#define COL_TILES (NHID / 16)        /* 4 */
#define TOT_TILES (ROW_TILES * COL_TILES)
#define GEMM_BLOCKS ((TOT_TILES + 7) / 8)

__global__ void __launch_bounds__(256)
gemm_bias(const float* __restrict__ A, const float* __restrict__ W,
          const float* __restrict__ bias, float* __restrict__ out) {
    const int wave = (blockIdx.x * blockDim.x + threadIdx.x) >> 5;
    if (wave >= TOT_TILES) return;           // whole wave exits uniformly
    const int rowTile = wave >> 2;
    const int colTile = wave & 3;
    const int lane = threadIdx.x & 31;
    const int ln = lane & 15;
    const int lh = lane >> 4;
    const int m0 = rowTile * 16;
    const int n0 = colTile * 16;
    const float* arow = A + (size_t)(m0 + ln) * NHID;

    v8f c = {0.f, 0.f, 0.f, 0.f, 0.f, 0.f, 0.f, 0.f};
#pragma unroll
    for (int s = 0; s < 16; s++) {
        const int k = 4 * s + 2 * lh;
        v2f a; a.x = arow[k];                 a.y = arow[k + 1];
        v2f b; b.x = W[(size_t)k * NHID + n0 + ln];
               b.y = W[(size_t)(k + 1) * NHID + n0 + ln];
        c = __builtin_amdgcn_wmma_f32_16x16x4_f32(false, a, false, b,
                                                  (short)0, c, false, false);
    }
    const float bv = bias[n0 + ln];
#pragma unroll
    for (int r = 0; r < 8; r++)
        out[(size_t)(m0 + r + 8 * lh) * NHID + n0 + ln] = c[r] + bv;
}

// out = relu( relu(z*scale + shift) @ W + bias )  (BN+ReLU fused into A load)
__global__ void __launch_bounds__(256)
gemm_bn_relu(const float* __restrict__ Z, const float* __restrict__ scale,
             const float* __restrict__ shift, const float* __restrict__ W,
             const float* __restrict__ bias, float* __restrict__ out) {
    const int wave = (blockIdx.x * blockDim.x + threadIdx.x) >> 5;
    if (wave >= TOT_TILES) return;
    const int rowTile = wave >> 2;
    const int colTile = wave & 3;
    const int lane = threadIdx.x & 31;
    const int ln = lane & 15;
    const int lh = lane >> 4;
    const int m0 = rowTile * 16;
    const int n0 = colTile * 16;
    const float* arow = Z + (size_t)(m0 + ln) * NHID;

    v8f c = {0.f, 0.f, 0.f, 0.f, 0.f, 0.f, 0.f, 0.f};
#pragma unroll
    for (int s = 0; s < 16; s++) {
        const int k = 4 * s + 2 * lh;
        v2f a;
        a.x = fmaxf(arow[k]     * scale[k]     + shift[k],     0.0f);
        a.y = fmaxf(arow[k + 1] * scale[k + 1] + shift[k + 1], 0.0f);
        v2f b; b.x = W[(size_t)k * NHID + n0 + ln];
               b.y = W[(size_t)(k + 1) * NHID + n0 + ln];
        c = __builtin_amdgcn_wmma_f32_16x16x4_f32(false, a, false, b,
                                                  (short)0, c, false, false);
    }
    const float bv = bias[n0 + ln];
#pragma unroll
    for (int r = 0; r < 8; r++)
        out[(size_t)(m0 + r + 8 * lh) * NHID + n0 + ln] =
            fmaxf(c[r] + bv, 0.0f);
}

// --------------------------------------------------------- BatchNorm stats
__global__ void __launch_bounds__(256)
bn_stats(const float* __restrict__ z, float* __restrict__ accum) {
    __shared__ float s_sum[256], s_sq[256];
    const int c = threadIdx.x & 63;
    const int g = threadIdx.x >> 6;      // 0..3
    float sum = 0.f, sq = 0.f;
    for (int row = blockIdx.x * 4 + g; row < NNODES; row += gridDim.x * 4) {
        float v = z[(size_t)row * NHID + c];
        sum += v; sq += v * v;
    }
    s_sum[threadIdx.x] = sum;
    s_sq[threadIdx.x] = sq;
    __syncthreads();
    if (g == 0) {
        sum = s_sum[c] + s_sum[c + 64] + s_sum[c + 128] + s_sum[c + 192];
        sq  = s_sq[c]  + s_sq[c + 64]  + s_sq[c + 128]  + s_sq[c + 192];
        atomicAdd(&accum[c], sum);
        atomicAdd(&accum[64 + c], sq);
    }
}

__global__ void bn_finalize(const float* __restrict__ accum,
                            const float* __restrict__ gamma,
                            const float* __restrict__ beta,
                            float* __restrict__ scale,
                            float* __restrict__ shift) {
    int c = threadIdx.x;
    float invN = 1.0f / (float)NNODES;
    float mu = accum[c] * invN;
    float var = accum[64 + c] * invN - mu * mu;   // biased var (jnp.var)
    float sc = gamma[c] * rsqrtf(var + BN_EPS);
    scale[c] = sc;
    shift[c] = beta[c] - mu * sc;
}

// -------------------------------------------------------------- readout
__global__ void count_kernel(const int* __restrict__ batch,
                             float* __restrict__ counts) {
    int i = blockIdx.x * blockDim.x + threadIdx.x;
    if (i < NNODES) atomicAdd(&counts[batch[i]], 1.0f);
}

__global__ void pool_kernel(const float* __restrict__ h,
                            const int* __restrict__ batch,
                            float* __restrict__ pooled, int off) {
    int tid = blockIdx.x * blockDim.x + threadIdx.x;
    int node = tid >> 4;
    int c = (tid & 15) * 4;
    if (node >= NNODES) return;
    int g = batch[node];
    float4 v = *(const float4*)(h + (size_t)node * NHID + c);
    float* p = pooled + (size_t)g * (3 * NHID) + off + c;
    atomicAdd(p + 0, v.x);
    atomicAdd(p + 1, v.y);
    atomicAdd(p + 2, v.z);
    atomicAdd(p + 3, v.w);
}

__global__ void final_kernel(const float* __restrict__ pooled,
                             const float* __restrict__ counts,
                             const float* __restrict__ linW,
                             const float* __restrict__ linb,
                             float* __restrict__ out) {
    int g = blockIdx.x;          // 64 graphs
    int j = threadIdx.x;         // 64 outputs
    float inv = 1.0f / fmaxf(counts[g], 1.0f);
    const float* pr = pooled + (size_t)g * (3 * NHID);
    float acc = 0.f;
#pragma unroll 4
    for (int k = 0; k < 3 * NHID; k++)
        acc += pr[k] * linW[(size_t)k * NHID + j];
    out[(size_t)g * NHID + j] = acc * inv + linb[j];
}

// ------------------------------------------------------------------- host
extern "C" void kernel_launch(void* const* d_in, const int* in_sizes, int n_in,
                              void* d_out, int out_size, void* d_ws, size_t ws_size,
                              hipStream_t stream) {
    const float* x = nullptr;
    const int* edge = nullptr;
    const int* batch = nullptr;
    const float* linW = nullptr;
    const float* w4096[8];
    int n4096 = 0;
    const float* v64[16];
    int n64 = 0;

    for (int i = 0; i < n_in; i++) {
        switch (in_sizes[i]) {
            case NNODES * NHID:  x    = (const float*)d_in[i]; break;  // 3.2M
            case 2 * NEDGES:     edge = (const int*)d_in[i];   break;  // 1.6M
            case NNODES:         batch = (const int*)d_in[i];  break;  // 50000
            case 3 * NHID * NHID: linW = (const float*)d_in[i]; break; // 12288
            case NHID * NHID:    if (n4096 < 8)  w4096[n4096++] = (const float*)d_in[i]; break;
            case NHID:           if (n64 < 16)   v64[n64++]     = (const float*)d_in[i]; break;
            default: break;
        }
    }
    // jax-sorted leaf order per conv: W1, W2, b1, b2, beta, gamma; lin_b last
    const float *W1[3], *W2[3], *b1[3], *b2[3], *beta[3], *gamma[3];
    for (int l = 0; l < 3; l++) {
        W1[l] = w4096[2 * l];
        W2[l] = w4096[2 * l + 1];
        b1[l] = v64[4 * l + 0];
        b2[l] = v64[4 * l + 1];
        beta[l] = v64[4 * l + 2];
        gamma[l] = v64[4 * l + 3];
    }
    const float* lin_b = v64[12];

    const int* e_src = edge;
    const int* e_dst = edge + NEDGES;

    // workspace layout
    float* t      = (float*)d_ws;          // [NF]
    float* z      = t + NF;                // [NF]
    float* h      = z + NF;                // [NF]
    float* accum  = h + NF;                // [128]
    float* scale  = accum + 128;           // [64]
    float* shiftv = scale + 64;            // [64]
    float* counts = shiftv + 64;           // [64]
    float* pooled = counts + 64;           // [64 * 192]

    // zero accumulators that persist across the whole launch
    zero_f32<<<32, 256, 0, stream>>>(counts, NGRAPH + NGRAPH * 3 * NHID);
    count_kernel<<<(NNODES + 255) / 256, 256, 0, stream>>>(batch, counts);

    const float* hin = x;
    for (int l = 0; l < 3; l++) {
        // t = h_in  (self term, eps = 0)
        copy_f32<<<(NF + 255) / 256, 256, 0, stream>>>(hin, t, NF);
        // t[dst] += h_in[src]
        edge_scatter<<<(NEDGES * 16 + 255) / 256, 256, 0, stream>>>(hin, e_src, e_dst, t);
        // z = t @ W1 + b1
        gemm_bias<<<GEMM_BLOCKS, 256, 0, stream>>>(t, W1[l], b1[l], z);
        // BN stats -> scale/shift
        zero_f32<<<1, 128, 0, stream>>>(accum, 128);
        bn_stats<<<128, 256, 0, stream>>>(z, accum);
        bn_finalize<<<1, 64, 0, stream>>>(accum, gamma[l], beta[l], scale, shiftv);
        // h = relu( relu(z*scale+shift) @ W2 + b2 )
        gemm_bn_relu<<<GEMM_BLOCKS, 256, 0, stream>>>(z, scale, shiftv, W2[l], b2[l], h);
        // pooled[:, l*64 : l*64+64] += per-graph sums of h
        pool_kernel<<<(NNODES * 16 + 255) / 256, 256, 0, stream>>>(h, batch, pooled, l * NHID);
        hin = h;   // stream-ordered: safe to reuse buffers each layer
    }

    final_kernel<<<NGRAPH, NHID, 0, stream>>>(pooled, counts, linW, lin_b, (float*)d_out);
}